// Generator1_56358560858126
// MI455X (gfx1250) — compile-verified
//
#include <hip/hip_runtime.h>
#include <hip/hip_bf16.h>

#define NN 5000
#define EE 80000
#define EPS 1e-3f

#define KSPLIT 10
#define KCHUNK (NN / KSPLIT)   // 500, multiple of 4

typedef __attribute__((ext_vector_type(2))) float v2f;
typedef __attribute__((ext_vector_type(8))) float v8f;

// ---------------- workspace layout (floats) ----------------
// sum1    : [N*35]            @ 0
// cnt     : [N]               @ 175000
// sum3    : [N*160]           @ 180000
// x1      : [N*35]            @ 980000
// x3      : [N*160]           @ 1155000
// partial : [KSPLIT*160*160]  @ 1955000   (split-K Gram partials)
#define OFF_SUM1 0
#define OFF_CNT  175000
#define OFF_SUM3 180000
#define OFF_X1   980000
#define OFF_X3   1155000
#define OFF_PART 1955000

__global__ void zero_kernel(float* __restrict__ p, int n) {
    int t = blockIdx.x * blockDim.x + threadIdx.x;
    if (t < n) p[t] = 0.0f;
}

// in-degree of dst (shared by both layers)
__global__ void degree_kernel(const long long* __restrict__ ei, float* __restrict__ cnt) {
    int e = blockIdx.x * blockDim.x + threadIdx.x;
    if (e >= EE) return;
    int dst = (int)ei[EE + e];
    atomicAdd(&cnt[dst], 1.0f);
}

// fused NNConv message kernel: thread = (edge, out_feature).
// theta_e[i,o] = relu(a_e * W[i*OUT+o] + b[i*OUT+o]) regenerated on the fly
// (W/b stay hot in WGP$/L2; avoids materializing E*IN*OUT theta = up to 1.8 GB).
template <int IN, int OUT>
__global__ void edge_msg_kernel(const float* __restrict__ xin,
                                const long long* __restrict__ ei,
                                const float* __restrict__ ea,
                                const float* __restrict__ W,
                                const float* __restrict__ b,
                                float* __restrict__ summ) {
    int t = blockIdx.x * blockDim.x + threadIdx.x;
    if (t >= EE * OUT) return;
    int e = t / OUT;
    int o = t - e * OUT;
    int src = (int)ei[e];
    int dst = (int)ei[EE + e];
    float a = ea[e];
    const float* __restrict__ xr = xin + src * IN;
    float acc = 0.0f;
#pragma unroll
    for (int i = 0; i < IN; ++i) {
        float th = fmaf(a, W[i * OUT + o], b[i * OUT + o]);   // a*W + b
        th = th > 0.0f ? th : 0.0f;                           // relu
        acc = fmaf(xr[i], th, acc);                           // dot with x[src]
    }
    atomicAdd(&summ[dst * OUT + o], acc);
}

// node update: mean-aggregate + x@root + bias -> BN(eval) -> sigmoid
template <int IN, int OUT>
__global__ void node_kernel(const float* __restrict__ xin,
                            const float* __restrict__ summ,
                            const float* __restrict__ cnt,
                            const float* __restrict__ root,   // [IN, OUT]
                            const float* __restrict__ bias,
                            const float* __restrict__ g,
                            const float* __restrict__ be,
                            const float* __restrict__ rm,
                            const float* __restrict__ rv,
                            float* __restrict__ xout) {
    int t = blockIdx.x * blockDim.x + threadIdx.x;
    if (t >= NN * OUT) return;
    int n = t / OUT;
    int o = t - n * OUT;
    float c = cnt[n];
    float mean = summ[n * OUT + o] / fmaxf(c, 1.0f);
    const float* __restrict__ xr = xin + n * IN;
    float r = 0.0f;
#pragma unroll
    for (int i = 0; i < IN; ++i) r = fmaf(xr[i], root[i * OUT + o], r);
    float h = mean + r + bias[o];
    float bn = (h - rm[o]) * (g[o] * __frsqrt_rn(rv[o] + EPS)) + be[o];
    xout[n * OUT + o] = 1.0f / (1.0f + __expf(-bn));
}

// Split-K Gram matrix: partial[s] = x3[s-th K chunk]^T @ x3[s-th K chunk]
// via V_WMMA_F32_16X16X4_F32. One wave (32 threads) per (tile, split);
// 10x10 tiles x KSPLIT splits = 1000 wave32s (vs 100 before -> 10x shorter
// dependent-WMMA critical path, much better WGP occupancy).
// f32 A 16x4 layout: lanes 0-15 -> {K0,K1}, lanes 16-31 -> {K2,K3}, M = lane&15.
// f32 B 4x16 layout: same striping with N = lane&15.
// C/D: VGPR r: lanes 0-15 -> (M=r, N=lane), lanes 16-31 -> (M=r+8, N=lane-16).
__global__ void gram_wmma_kernel(const float* __restrict__ x3,
                                 float* __restrict__ partial) {
    int tile  = blockIdx.x % 100;     // 0..99
    int split = blockIdx.x / 100;     // 0..KSPLIT-1
    int tm = (tile / 10) * 16;        // output row block (feature)
    int tn = (tile % 10) * 16;        // output col block (feature)
    int lane = threadIdx.x;           // 0..31, EXEC all ones (WMMA requirement)
    int half = lane >> 4;             // 0: K pair {0,1}, 1: K pair {2,3}
    int l = lane & 15;

    int k0 = split * KCHUNK;
    v8f c = {};
    for (int k = k0; k < k0 + KCHUNK; k += 4) {
        int r0 = k + 2 * half;
        int r1 = r0 + 1;
        v2f a, b;
        a.x = x3[r0 * 160 + tm + l];   // A[m, kk] = x3[k, m]
        a.y = x3[r1 * 160 + tm + l];
        b.x = x3[r0 * 160 + tn + l];   // B[kk, n] = x3[k, n]
        b.y = x3[r1 * 160 + tn + l];
        c = __builtin_amdgcn_wmma_f32_16x16x4_f32(
                /*neg_a=*/false, a, /*neg_b=*/false, b,
                /*c_mod=*/(short)0, c, /*reuse_a=*/false, /*reuse_b=*/false);
    }
    float* pp = partial + split * (160 * 160);
#pragma unroll
    for (int r = 0; r < 8; ++r) {
        int M = r + 8 * half;
        pp[(tm + M) * 160 + tn + l] = c[r];
    }
}

// deterministic fixed-order reduction of the split-K partials -> d_out
__global__ void gram_reduce_kernel(const float* __restrict__ partial,
                                   float* __restrict__ out) {
    int t = blockIdx.x * blockDim.x + threadIdx.x;
    if (t >= 160 * 160) return;
    float s = 0.0f;
#pragma unroll
    for (int i = 0; i < KSPLIT; ++i) s += partial[i * (160 * 160) + t];
    out[t] = s;
}

extern "C" void kernel_launch(void* const* d_in, const int* in_sizes, int n_in,
                              void* d_out, int out_size, void* d_ws, size_t ws_size,
                              hipStream_t stream) {
    const float*     x     = (const float*)d_in[0];
    const long long* ei    = (const long long*)d_in[1];   // int64 edge_index [2,E]
    const float*     ea    = (const float*)d_in[2];
    const float*     W1    = (const float*)d_in[3];
    const float*     b1    = (const float*)d_in[4];
    const float*     root1 = (const float*)d_in[5];
    const float*     bias1 = (const float*)d_in[6];
    const float*     g1    = (const float*)d_in[7];
    const float*     be1   = (const float*)d_in[8];
    const float*     rm1   = (const float*)d_in[9];
    const float*     rv1   = (const float*)d_in[10];
    const float*     W3    = (const float*)d_in[11];
    const float*     b3    = (const float*)d_in[12];
    const float*     root3 = (const float*)d_in[13];
    const float*     bias3 = (const float*)d_in[14];
    const float*     g3    = (const float*)d_in[15];
    const float*     be3   = (const float*)d_in[16];
    const float*     rm3   = (const float*)d_in[17];
    const float*     rv3   = (const float*)d_in[18];

    float* ws      = (float*)d_ws;
    float* sum1    = ws + OFF_SUM1;
    float* cnt     = ws + OFF_CNT;
    float* sum3    = ws + OFF_SUM3;
    float* x1      = ws + OFF_X1;
    float* x3      = ws + OFF_X3;
    float* partial = ws + OFF_PART;
    float* out     = (float*)d_out;

    const int B = 256;

    // 1) zero accumulators (sum1, cnt, sum3 are contiguous: [0, 980000))
    {
        int n = OFF_X1;
        zero_kernel<<<(n + B - 1) / B, B, 0, stream>>>(ws, n);
    }
    // 2) in-degree
    degree_kernel<<<(EE + B - 1) / B, B, 0, stream>>>(ei, cnt);
    // 3) layer-1 messages (E*35 threads)
    {
        int n = EE * 35;
        edge_msg_kernel<35, 35><<<(n + B - 1) / B, B, 0, stream>>>(x, ei, ea, W1, b1, sum1);
    }
    // 4) layer-1 node update -> x1
    {
        int n = NN * 35;
        node_kernel<35, 35><<<(n + B - 1) / B, B, 0, stream>>>(
            x, sum1, cnt, root1, bias1, g1, be1, rm1, rv1, x1);
    }
    // 5) layer-2 messages (E*160 threads)
    {
        int n = EE * 160;
        edge_msg_kernel<35, 160><<<(n + B - 1) / B, B, 0, stream>>>(x1, ei, ea, W3, b3, sum3);
    }
    // 6) layer-2 node update -> x3
    {
        int n = NN * 160;
        node_kernel<35, 160><<<(n + B - 1) / B, B, 0, stream>>>(
            x1, sum3, cnt, root3, bias3, g3, be3, rm3, rv3, x3);
    }
    // 7) split-K Gram via f32 WMMA: 100 tiles x KSPLIT chunks, one wave32 each
    gram_wmma_kernel<<<100 * KSPLIT, 32, 0, stream>>>(x3, partial);
    // 8) deterministic fixed-order reduction -> d_out [160,160]
    {
        int n = 160 * 160;
        gram_reduce_kernel<<<(n + B - 1) / B, B, 0, stream>>>(partial, out);
    }
}